// ReduceBoundingBoxes_32031866094312
// MI455X (gfx1250) — compile-verified
//
#include <hip/hip_runtime.h>
#include <stdint.h>

#define P_ 96
#define N_ (P_ * P_)            // 9216
#define PROB_THR_F 0.9f
#define IOU_THR_F  0.5f
#define XPS_F 16.0f
#define YPS_F 16.0f
#define W_F 1536.0f
#define H_F 1536.0f
#define KEEP_CAP 1024           // LDS-cached kept boxes (overflow -> global via index)

typedef __attribute__((ext_vector_type(4))) unsigned int u32x4_t;
typedef __attribute__((ext_vector_type(8))) int          i32x8_t;
typedef __attribute__((ext_vector_type(4))) int          i32x4_t;

// ---------------------------------------------------------------------------
// Workspace layout (float units):
//   [0,N)        score[n]
//   [N,5N)       bbox4[n]  = float4{x1,y1,x2,y2}   (AoS, 16B records)
//   [5N,6N)      s_sc[rank]                         (sorted scores)
//   [6N,10N)     s_box4[rank] = float4{x1,y1,x2,y2} (sorted boxes)
//   [10N]        int valid-count
// ---------------------------------------------------------------------------

__global__ void bb_zero_kernel(float* __restrict__ out, int out_size,
                               int* __restrict__ count) {
  int t = blockIdx.x * blockDim.x + threadIdx.x;
  if (t < out_size) out[t] = 0.0f;
  if (t == 0) *count = 0;
}

__global__ void bb_decode_kernel(const float* __restrict__ x,
                                 float* __restrict__ ws,
                                 int* __restrict__ count) {
  int n = blockIdx.x * blockDim.x + threadIdx.x;   // n in [0, N_)
  int i = n / P_;
  int j = n - i * P_;
  float sc = x[n];
  // Explicit round-to-nearest mul/add: no FMA contraction, matches the
  // reference's separate mul-then-add before rounding.
  float bx = __fadd_rn(__fmul_rn(x[N_ + n],     XPS_F), (float)i * XPS_F);
  float by = __fadd_rn(__fmul_rn(x[2 * N_ + n], YPS_F), (float)j * YPS_F);
  float bw = __fmul_rn(x[3 * N_ + n], W_F);
  float bh = __fmul_rn(x[4 * N_ + n], H_F);
  float4 b;
  b.x = __builtin_rintf(bx);                       // round half to even
  b.y = __builtin_rintf(by);
  b.z = __builtin_rintf(__fadd_rn(bw, bx));
  b.w = __builtin_rintf(__fadd_rn(bh, by));
  ws[n] = sc;
  reinterpret_cast<float4*>(ws + N_)[n] = b;       // one b128 store
  // Count valid boxes: one ballot+popc per wave32, one atomic per wave.
  unsigned long long m = __ballot(sc > PROB_THR_F);
  if ((threadIdx.x & 31u) == 0u) atomicAdd(count, __popcll(m));
}

// --- Tensor Data Mover: 1D global->LDS copy of nelem 4-byte words ----------
__device__ __forceinline__ void tdm_copy_to_lds_1d(uint32_t lds_off,
                                                   const void* gptr,
                                                   uint32_t nelem) {
  uint64_t ga = (uint64_t)(uintptr_t)gptr;
  u32x4_t g0;
  g0[0] = 1u;                                   // count=1 valid descriptor
  g0[1] = lds_off;                              // lds_addr (bytes)
  g0[2] = (uint32_t)ga;                         // global_addr[31:0]
  g0[3] = (uint32_t)((ga >> 32) & 0x1FFFFFFull) // global_addr[56:32]
        | (2u << 30);                           // type = 2 ("image")
  i32x8_t g1;
  g1[0] = (int)(2u << 16);                            // data_size = 4B
  g1[1] = (int)((nelem & 0xFFFFu) << 16);             // tensor_dim0 lo16
  g1[2] = (int)(((nelem >> 16) & 0xFFFFu) | (1u << 16)); // dim0 hi | tensor_dim1=1
  g1[3] = (int)((nelem & 0xFFFFu) << 16);             // tile_dim0 = nelem
  g1[4] = 1;                                          // tile_dim1 = 1, tile_dim2 = 0
  g1[5] = (int)nelem;                                 // tensor_dim0_stride lo32
  g1[6] = (int)((nelem & 0xFFFFu) << 16);             // stride0 hi=0 | stride1 lo16
  g1[7] = (int)(nelem >> 16);                         // stride1 hi
  i32x4_t gz4; gz4[0] = 0; gz4[1] = 0; gz4[2] = 0; gz4[3] = 0;
  i32x8_t gz8;
  gz8[0] = 0; gz8[1] = 0; gz8[2] = 0; gz8[3] = 0;
  gz8[4] = 0; gz8[5] = 0; gz8[6] = 0; gz8[7] = 0;
  __builtin_amdgcn_tensor_load_to_lds(g0, g1, gz4, gz4, gz8, 0);
  __builtin_amdgcn_s_wait_tensorcnt(0);
}

// Rank = #{valid j ahead in stable descending sort}. A score greater than a
// valid score is itself > 0.9, so raw-score comparison is sufficient.
__global__ void bb_rank_kernel(float* __restrict__ ws) {
  __shared__ float ssc[N_];                     // 36 KB score table (TDM-staged)
  if (threadIdx.x == 0) {                       // one wave issues the TDM op
    tdm_copy_to_lds_1d((uint32_t)(uintptr_t)&ssc[0], ws, (uint32_t)N_);
  }
  __syncthreads();
  int i = blockIdx.x * blockDim.x + threadIdx.x;
  float s = ssc[i];
  if (s > PROB_THR_F) {
    const float4* ssc4 = reinterpret_cast<const float4*>(ssc);
    int rank = 0;
#pragma unroll 4
    for (int j4 = 0; j4 < N_ / 4; ++j4) {       // b128 LDS broadcast reads
      float4 sj = ssc4[j4];
      int j = j4 * 4;
      rank += (int)((sj.x > s) || ((sj.x == s) && (j + 0 < i)));
      rank += (int)((sj.y > s) || ((sj.y == s) && (j + 1 < i)));
      rank += (int)((sj.z > s) || ((sj.z == s) && (j + 2 < i)));
      rank += (int)((sj.w > s) || ((sj.w == s) && (j + 3 < i)));
    }
    ws[5 * N_ + rank] = s;
    reinterpret_cast<float4*>(ws + 6 * N_)[rank] =
        reinterpret_cast<const float4*>(ws + N_)[i];
  }
}

// Sequential greedy NMS on a single wave32. Cross-lane vote via ballot;
// __syncthreads is a cheap single-wave LDS ordering point. Kept boxes are
// appended to an LDS float4 cache; box i+1 is software-pipelined.
__global__ void bb_nms_kernel(const float* __restrict__ ws,
                              const int* __restrict__ count,
                              float* __restrict__ out) {
  __shared__ float4 kbox[KEEP_CAP];             // 16 KB kept-box cache
  __shared__ int    kidx[N_];                   // 36 KB kept sorted-indices
  const float*  s_sc   = ws + 5 * N_;
  const float4* s_box4 = reinterpret_cast<const float4*>(ws + 6 * N_);
  const int K = *count;
  const int lane = (int)threadIdx.x;
  int kept = 0;
  float4 b;
  if (K > 0) b = s_box4[0];
  for (int i = 0; i < K; ++i) {
    // Start fetching the next box now; latency overlaps this iteration.
    float4 bnext;
    if (i + 1 < K) bnext = s_box4[i + 1];
    float areai = (b.z - b.x) * (b.w - b.y);
    int sup = 0;
    for (int m = lane; m < kept; m += 32) {
      float4 c;
      if (m < KEEP_CAP) c = kbox[m];            // one ds_load_b128
      else              c = s_box4[kidx[m]];
      float iw = fmaxf(fminf(b.z, c.z) - fmaxf(b.x, c.x), 0.0f);
      float ih = fmaxf(fminf(b.w, c.w) - fmaxf(b.y, c.y), 0.0f);
      float inter = iw * ih;
      float uni = areai + (c.z - c.x) * (c.w - c.y) - inter;
      sup |= (uni > 0.0f) && ((inter / uni) > IOU_THR_F);   // match reference
    }
    if (__ballot(sup) == 0ull) {                // box i survives
      if (lane == 0) {
        kidx[kept] = i;
        if (kept < KEEP_CAP) kbox[kept] = b;
        out[i * 5 + 0] = s_sc[i];
      } else if (lane == 1) out[i * 5 + 1] = b.x;
      else if (lane == 2)   out[i * 5 + 2] = b.y;
      else if (lane == 3)   out[i * 5 + 3] = b.z - b.x;
      else if (lane == 4)   out[i * 5 + 4] = b.w - b.y;
      else if (lane == 5)   out[5 * N_ + i] = 1.0f;   // keeps flag
      kept++;
    }
    __syncthreads();                            // LDS write->read ordering
    b = bnext;
  }
}

extern "C" void kernel_launch(void* const* d_in, const int* in_sizes, int n_in,
                              void* d_out, int out_size, void* d_ws, size_t ws_size,
                              hipStream_t stream) {
  (void)in_sizes; (void)n_in; (void)ws_size;
  const float* x = (const float*)d_in[0];
  float* out = (float*)d_out;
  float* ws  = (float*)d_ws;
  int* count = (int*)(ws + 10 * N_);

  int zb = (out_size + 255) / 256;
  bb_zero_kernel<<<zb, 256, 0, stream>>>(out, out_size, count);
  bb_decode_kernel<<<N_ / 256, 256, 0, stream>>>(x, ws, count);
  bb_rank_kernel<<<N_ / 256, 256, 0, stream>>>(ws);
  bb_nms_kernel<<<1, 32, 0, stream>>>(ws, count, out);
}